// CustomLossWithCovariance_1675037245673
// MI455X (gfx1250) — compile-verified
//
#include <hip/hip_runtime.h>
#include <hip/hip_bf16.h>

// CDNA5 / gfx1250: wave32, WMMA 16x16x4 f32.
typedef float v2f __attribute__((ext_vector_type(2)));
typedef float v8f __attribute__((ext_vector_type(8)));

#define BLOCK 256
#define WAVES_PER_BLOCK (BLOCK / 32)
#define GRID 2048
#define SAMPLES_PER_TILE 20   // 5 diagonal 3x3 blocks x K=4 samples per V_WMMA_F32_16X16X4_F32
#define UNROLL 4              // tiles per wave iteration (independent accumulators)

__global__ __launch_bounds__(64) void zero_ws_kernel(double* ws) {
    if (blockIdx.x == 0 && threadIdx.x < 6) ws[threadIdx.x] = 0.0;
}

// Streams predictions/targets once (non-temporal), accumulates the 3x3 Gram
// G = D^T D via v_wmma_f32_16x16x4_f32, packing 5 disjoint 3x3 diagonal
// blocks (20 samples) per WMMA. A(16x4) and B(4x16)=A^T share the same lane
// layout (K split across lane halves), so one operand register pair feeds both.
// Scalar (wave-uniform) loop index via readfirstlane -> SALU loop control,
// GVS addressing (saddr + 32-bit lane offset + imm), 16 loads in flight.
__global__ __launch_bounds__(BLOCK) void gram_wmma_kernel(
        const float* __restrict__ pred,
        const float* __restrict__ targ,
        double* __restrict__ ws,
        int numGroups) {
    __shared__ float lds[WAVES_PER_BLOCK * 256];

    const int lane   = threadIdx.x & 31;
    const int waveId = threadIdx.x >> 5;
    const int M  = lane & 15;       // matrix row (A) / col (B) this lane feeds
    const int hk = lane >> 4;       // lane half selects K pair {0,1} or {2,3}

    int blk  = M / 3;  blk  = (blk  > 4) ? 4 : blk;       // which 3x3 diagonal block
    int comp = M - 3 * blk; comp = (comp > 2) ? 2 : comp; // vector component 0..2
    const int K0 = 2 * hk;
    const int K1 = 2 * hk + 1;
    // per-lane 32-bit float offsets inside a 20-sample (60-float) tile
    const int off0 = 12 * blk + 3 * K0 + comp;
    const int off1 = 12 * blk + 3 * K1 + comp;

    v8f acc[UNROLL];
#pragma unroll
    for (int u = 0; u < UNROLL; ++u) acc[u] = (v8f){};

    // wave-uniform scalar loop bounds (keeps loop control on SALU, EXEC all-ones)
    const int waveGlobal = __builtin_amdgcn_readfirstlane(
            (int)(blockIdx.x * WAVES_PER_BLOCK + waveId));
    const int totalWaves = (int)(gridDim.x * WAVES_PER_BLOCK);

    for (int g = waveGlobal; g < numGroups; g += totalWaves) {
        const float* __restrict__ p = pred + (long)g * (3L * SAMPLES_PER_TILE * UNROLL);
        const float* __restrict__ t = targ + (long)g * (3L * SAMPLES_PER_TILE * UNROLL);
#pragma unroll
        for (int u = 0; u < UNROLL; ++u) {
            const int uo = u * (3 * SAMPLES_PER_TILE);
            // streaming data: non-temporal so we don't thrash L2
            float p0 = __builtin_nontemporal_load(p + off0 + uo);
            float p1 = __builtin_nontemporal_load(p + off1 + uo);
            float t0 = __builtin_nontemporal_load(t + off0 + uo);
            float t1 = __builtin_nontemporal_load(t + off1 + uo);
            v2f a;
            a.x = p0 - t0;   // K = 2*half
            a.y = p1 - t1;   // K = 2*half + 1
            // D = A*B + C, B == A (Gram). 8 args: neg_a, A, neg_b, B, c_mod, C, reuse_a, reuse_b
            acc[u] = __builtin_amdgcn_wmma_f32_16x16x4_f32(
                    false, a, false, a, (short)0, acc[u], false, false);
        }
    }

    // combine the independent accumulators
#pragma unroll
    for (int u = 1; u < UNROLL; ++u)
#pragma unroll
        for (int v = 0; v < 8; ++v) acc[0][v] += acc[u][v];

    // Spill C to LDS: VGPR v holds rows v (lanes 0-15, N=lane) and v+8 (lanes 16-31, N=lane-16)
#pragma unroll
    for (int v = 0; v < 8; ++v)
        lds[waveId * 256 + v * 32 + lane] = acc[0][v];
    __syncthreads();

    // 6 threads gather the 6 unique Gram entries across 8 waves x 5 diagonal blocks
    if (threadIdx.x < 6) {
        const int gi[6] = {0, 1, 2, 0, 0, 1};
        const int gj[6] = {0, 1, 2, 1, 2, 2};
        const int i = gi[threadIdx.x], j = gj[threadIdx.x];
        float s = 0.0f;
        for (int w = 0; w < WAVES_PER_BLOCK; ++w) {
            for (int b = 0; b < 5; ++b) {
                const int r = 3 * b + i;
                const int c = 3 * b + j;
                const int idx = w * 256 + (r & 7) * 32 + ((r >> 3) ? 16 : 0) + c;
                s += lds[idx];
            }
        }
        atomicAdd(&ws[threadIdx.x], (double)s);   // global_atomic_add_f64
    }
}

__global__ __launch_bounds__(32) void finalize_kernel(
        const float* __restrict__ pred,
        const float* __restrict__ targ,
        const float* __restrict__ sigma,
        const double* __restrict__ ws,
        float* __restrict__ out,
        long startSample, long B) {
    if (threadIdx.x != 0 || blockIdx.x != 0) return;

    double g[6];
    for (int k = 0; k < 6; ++k) g[k] = ws[k];

    // tail samples not covered by the tiled WMMA kernel
    for (long s = startSample; s < B; ++s) {
        const double d0 = (double)pred[3 * s + 0] - (double)targ[3 * s + 0];
        const double d1 = (double)pred[3 * s + 1] - (double)targ[3 * s + 1];
        const double d2 = (double)pred[3 * s + 2] - (double)targ[3 * s + 2];
        g[0] += d0 * d0; g[1] += d1 * d1; g[2] += d2 * d2;
        g[3] += d0 * d1; g[4] += d0 * d2; g[5] += d1 * d2;
    }

    const double s00 = sigma[0], s01 = sigma[1], s02 = sigma[2];
    const double s10 = sigma[3], s11 = sigma[4], s12 = sigma[5];
    const double s20 = sigma[6], s21 = sigma[7], s22 = sigma[8];

    const double c00 = s11 * s22 - s12 * s21;
    const double c01 = s10 * s22 - s12 * s20;
    const double c02 = s10 * s21 - s11 * s20;
    const double det = s00 * c00 - s01 * c01 + s02 * c02;
    const double inv = 1.0 / det;

    const double i00 =  c00 * inv;
    const double i01 = (s02 * s21 - s01 * s22) * inv;
    const double i02 = (s01 * s12 - s02 * s11) * inv;
    const double i10 = -c01 * inv;
    const double i11 = (s00 * s22 - s02 * s20) * inv;
    const double i12 = (s02 * s10 - s00 * s12) * inv;
    const double i20 =  c02 * inv;
    const double i21 = (s01 * s20 - s00 * s21) * inv;
    const double i22 = (s00 * s11 - s01 * s10) * inv;

    // trace(Sigma^{-1} * G), G symmetric
    const double msum = i00 * g[0] + i11 * g[1] + i22 * g[2]
                      + (i01 + i10) * g[3] + (i02 + i20) * g[4] + (i12 + i21) * g[5];

    const double logdet = log(fabs(det));   // SPD: det > 0
    const double loss = fabs(logdet + msum / (double)B);
    out[0] = (float)loss;
}

extern "C" void kernel_launch(void* const* d_in, const int* in_sizes, int n_in,
                              void* d_out, int out_size, void* d_ws, size_t ws_size,
                              hipStream_t stream) {
    const float* pred  = (const float*)d_in[0];
    const float* targ  = (const float*)d_in[1];
    const float* sigma = (const float*)d_in[2];
    float* out = (float*)d_out;
    double* ws = (double*)d_ws;

    const long B = (long)in_sizes[0] / 3;
    const long numTiles  = B / SAMPLES_PER_TILE;
    const int  numGroups = (int)(numTiles / UNROLL);
    const long startSample = (long)numGroups * (SAMPLES_PER_TILE * UNROLL);

    zero_ws_kernel<<<1, 64, 0, stream>>>(ws);
    gram_wmma_kernel<<<GRID, BLOCK, 0, stream>>>(pred, targ, ws, numGroups);
    finalize_kernel<<<1, 32, 0, stream>>>(pred, targ, sigma, ws, out, startSample, B);
}